// sample_and_group_32298154066786
// MI455X (gfx1250) — compile-verified
//
#include <hip/hip_runtime.h>
#include <hip/hip_bf16.h>

// PointNet++ sample-and-group for MI455X (gfx1250, wave32).
//
// Stage 1 (FPS):   one 1024-thread block per batch; each thread keeps its 16
//                  points (and min-dists) in VGPRS -- no per-step LDS sweep.
//                  The argmax reduction carries (dist, idx, x, y, z) so the
//                  winner's coordinates are broadcast without any lookup.
//                  2 barriers per step; ~1KB LDS.
// Stage 2 (ball):  V_WMMA_F32_16X16X4_F32 computes 16x16 squared-distance
//                  tiles (d = q^2 - 2 q.p + p^2 folded into A/B/C). Two tiles
//                  (32 candidates) per loop iteration -> one 32-bit ordered
//                  hit-mask per query row; ballot-driven ordered append gives
//                  "first 32 indices within radius" = reference top_k-of-index
//                  semantics; uniform early exit (~6% of candidates scanned).
// Stage 3 (group): pure gather + coalesced store; inputs L2-resident (33 MB of
//                  192 MB L2), output 68.7 MB -> ~5 us at 23.3 TB/s.

#define NBATCH  8
#define NPTS    16384
#define NCH     64
#define NPOINT  1024
#define NSAMPLE 32
#define RADIUS2 0.04f

typedef __attribute__((ext_vector_type(2))) float v2f;
typedef __attribute__((ext_vector_type(8))) float v8f;

// ---------------------------------------------------------------------------
// Kernel 1: furthest point sampling. grid = NBATCH, block = 1024 (32 waves).
// Writes new_xyz[b][c][m] directly (c-major as in reference output).
// ---------------------------------------------------------------------------
__global__ __launch_bounds__(1024) void fps_kernel(const float* __restrict__ xyz,
                                                   float* __restrict__ newxyz) {
  const int b    = blockIdx.x;
  const int tid  = threadIdx.x;
  const int lane = tid & 31;
  const int wave = tid >> 5;
  const float* xb = xyz + (size_t)b * 3 * NPTS;

  __shared__ float rd[32];
  __shared__ int   rn[32];
  __shared__ float rx[32], ry[32], rz[32];
  __shared__ float bcx, bcy, bcz;

  // Coordinates and min-dists live in VGPRs: n = i*1024 + tid (coalesced).
  float cx[16], cy[16], cz[16], mind[16];
#pragma unroll
  for (int i = 0; i < 16; ++i) {
    const int n = i * 1024 + tid;
    cx[i] = xb[n];
    cy[i] = xb[NPTS + n];
    cz[i] = xb[2 * NPTS + n];
    mind[i] = 1e10f;
  }
  // Seed the broadcast with point 0 (first FPS selection is index 0).
  if (tid == 0) { bcx = cx[0]; bcy = cy[0]; bcz = cz[0]; }
  __syncthreads();

  for (int m = 0; m < NPOINT; ++m) {
    const float lx = bcx, ly = bcy, lz = bcz;
    if (tid == 0) {
      newxyz[(size_t)b * 3 * NPOINT + 0 * NPOINT + m] = lx;
      newxyz[(size_t)b * 3 * NPOINT + 1 * NPOINT + m] = ly;
      newxyz[(size_t)b * 3 * NPOINT + 2 * NPOINT + m] = lz;
    }

    // Per-thread min-dist update + local argmax (strict '>' keeps the
    // first/smallest-index max, matching jnp.argmax semantics).
    float bd = -1.0f;
    int   bn = 0;
#pragma unroll
    for (int i = 0; i < 16; ++i) {
      const float dx = cx[i] - lx;
      const float dy = cy[i] - ly;
      const float dz = cz[i] - lz;
      const float d  = dx * dx + dy * dy + dz * dz;
      const float md = fminf(mind[i], d);
      mind[i] = md;
      if (md > bd) { bd = md; bn = i * 1024 + tid; }
    }
    // Recover this thread's best-point coordinates (slot = bn >> 10).
    const int bi = bn >> 10;
    float bx = cx[0], by = cy[0], bz = cz[0];
#pragma unroll
    for (int i = 1; i < 16; ++i) {
      if (bi == i) { bx = cx[i]; by = cy[i]; bz = cz[i]; }
    }

    // Wave32 argmax reduction carrying the winner's coordinates.
#pragma unroll
    for (int off = 16; off > 0; off >>= 1) {
      const float od = __shfl_xor(bd, off, 32);
      const int   on = __shfl_xor(bn, off, 32);
      const float ox = __shfl_xor(bx, off, 32);
      const float oy = __shfl_xor(by, off, 32);
      const float oz = __shfl_xor(bz, off, 32);
      if (od > bd || (od == bd && on < bn)) {
        bd = od; bn = on; bx = ox; by = oy; bz = oz;
      }
    }
    if (lane == 0) {
      rd[wave] = bd; rn[wave] = bn; rx[wave] = bx; ry[wave] = by; rz[wave] = bz;
    }
    __syncthreads();
    if (wave == 0) {
      bd = rd[lane]; bn = rn[lane]; bx = rx[lane]; by = ry[lane]; bz = rz[lane];
#pragma unroll
      for (int off = 16; off > 0; off >>= 1) {
        const float od = __shfl_xor(bd, off, 32);
        const int   on = __shfl_xor(bn, off, 32);
        const float ox = __shfl_xor(bx, off, 32);
        const float oy = __shfl_xor(by, off, 32);
        const float oz = __shfl_xor(bz, off, 32);
        if (od > bd || (od == bd && on < bn)) {
          bd = od; bn = on; bx = ox; by = oy; bz = oz;
        }
      }
      if (lane == 0) { bcx = bx; bcy = by; bcz = bz; }
    }
    __syncthreads();
    // (reads of bc* at the next loop top are ordered before the next write
    //  of bc* by the first barrier of the next iteration)
  }
}

// ---------------------------------------------------------------------------
// Kernel 2: ball query via WMMA. grid = NBATCH*64 blocks, block = 32 (1 wave).
// One wave owns 16 query points and sweeps candidates 32 at a time (two
// V_WMMA_F32_16X16X4_F32 tiles per iteration):
//   A[m] = (qx, qy, qz, q^2)            (16x4, loop-invariant)
//   B[:,n] = (-2px, -2py, -2pz, 1)      (4x16, per tile)
//   C[m][n] = p_n^2  (column-constant, matches C layout N = lane%16)
//   D = q^2 - 2 q.p + p^2 (same formulation as the reference).
// Layouts: A per ISA table (lane<16: K0/K1, lane>=16: K2/K3, lane=M);
//          B as the lane-transposed mirror (lane%16 = N, halves = K pairs);
//          C/D per documented 16x16 f32 layout.
// ---------------------------------------------------------------------------
__global__ __launch_bounds__(32) void ballquery_kernel(const float* __restrict__ xyz,
                                                       const float* __restrict__ newxyz,
                                                       int* __restrict__ wsidx) {
  const int b    = blockIdx.x >> 6;
  const int m0   = (blockIdx.x & 63) << 4;
  const int lane = threadIdx.x;
  const int ln   = lane & 15;
  const bool hi  = lane >= 16;
  const float* xb = xyz + (size_t)b * 3 * NPTS;
  const float* qb = newxyz + (size_t)b * 3 * NPOINT;

  __shared__ int sidx[16][NSAMPLE];

  // Build A (loop-invariant).
  const float qx = qb[m0 + ln];
  const float qy = qb[NPOINT + m0 + ln];
  const float qz = qb[2 * NPOINT + m0 + ln];
  const float q2 = qx * qx + qy * qy + qz * qz;
  v2f a;
  a.x = hi ? qz : qx;
  a.y = hi ? q2 : qy;

  int cnt = 0;  // lanes 0..15 own rows 0..15; lanes 16..31 idle for appends

  for (int n0 = 0; n0 < NPTS; n0 += 32) {
    // ---- tile 0: candidates n0 .. n0+15 ----
    const int na = n0 + ln;
    const float pxa = xb[na];
    const float pya = xb[NPTS + na];
    const float pza = xb[2 * NPTS + na];
    __builtin_prefetch(xb + na + 512, 0, 3);              // global_prefetch_b8
    __builtin_prefetch(xb + 2 * NPTS + na + 512, 0, 3);
    const float p2a = pxa * pxa + pya * pya + pza * pza;
    v2f ba;
    ba.x = hi ? (-2.0f * pza) : (-2.0f * pxa);
    ba.y = hi ? 1.0f          : (-2.0f * pya);
    v8f ca;
#pragma unroll
    for (int v = 0; v < 8; ++v) ca[v] = p2a;
    v8f da = __builtin_amdgcn_wmma_f32_16x16x4_f32(false, a, false, ba,
                                                   (short)0, ca, false, false);

    // ---- tile 1: candidates n0+16 .. n0+31 ----
    const int nb = n0 + 16 + ln;
    const float pxb = xb[nb];
    const float pyb = xb[NPTS + nb];
    const float pzb = xb[2 * NPTS + nb];
    const float p2b = pxb * pxb + pyb * pyb + pzb * pzb;
    v2f bb;
    bb.x = hi ? (-2.0f * pzb) : (-2.0f * pxb);
    bb.y = hi ? 1.0f          : (-2.0f * pyb);
    v8f cb;
#pragma unroll
    for (int v = 0; v < 8; ++v) cb[v] = p2b;
    v8f db = __builtin_amdgcn_wmma_f32_16x16x4_f32(false, a, false, bb,
                                                   (short)0, cb, false, false);

    // Ballot per accumulator VGPR: bits [15:0] = row v, bits [31:16] = row v+8
    // (cols = this tile's 16 candidates). Redistribute so lane r holds the
    // 32-candidate ordered hit mask of row r: bit p <-> candidate n0+p.
    unsigned mymask = 0u;
#pragma unroll
    for (int v = 0; v < 8; ++v) {
      const unsigned mka = __builtin_amdgcn_ballot_w32(da[v] <= RADIUS2);
      const unsigned mkb = __builtin_amdgcn_ballot_w32(db[v] <= RADIUS2);
      mymask |= (lane == v)     ? ((mka & 0xffffu) | (mkb << 16))          : 0u;
      mymask |= (lane == v + 8) ? ((mka >> 16) | (mkb & 0xffff0000u))      : 0u;
    }

    // Ordered append (ascending candidate index == reference's top_k of
    // smallest indices within radius).
    while (mymask && cnt < NSAMPLE) {
      const int bit = __builtin_ctz(mymask);
      sidx[lane][cnt] = n0 + bit;
      ++cnt;
      mymask &= mymask - 1u;
    }

    // Uniform early exit once all 16 rows are full.
    const unsigned full = __builtin_amdgcn_ballot_w32(cnt >= NSAMPLE || hi);
    if (full == 0xffffffffu) break;
  }

  __syncthreads();
  if (lane < 16) {
    if (cnt == 0) { sidx[lane][0] = 0; cnt = 1; }   // unreachable (self-hit), safety
    const int f = sidx[lane][0];
    for (int k = cnt; k < NSAMPLE; ++k) sidx[lane][k] = f;  // pad with first hit
  }
  __syncthreads();

#pragma unroll
  for (int r = 0; r < 16; ++r) {
    wsidx[(size_t)(b * NPOINT + m0 + r) * NSAMPLE + lane] = sidx[r][lane];
  }
}

// ---------------------------------------------------------------------------
// Kernel 3: gather + group. grid = NBATCH*NPOINT blocks (one per (b,m)),
// block = 256. Output new_features[b][c][m][s]; c<3 is grouped_xyz - new_xyz,
// c>=3 is grouped features. lane = s -> fully coalesced 128B stores.
// ---------------------------------------------------------------------------
__global__ __launch_bounds__(256) void group_kernel(const float* __restrict__ xyz,
                                                    const float* __restrict__ features,
                                                    const float* __restrict__ newxyz,
                                                    const int* __restrict__ wsidx,
                                                    float* __restrict__ outfeat) {
  const int b   = blockIdx.x >> 10;
  const int m   = blockIdx.x & 1023;
  const int tid = threadIdx.x;

  __shared__ int   sidx[NSAMPLE];
  __shared__ float sq[3];
  if (tid < NSAMPLE) sidx[tid] = wsidx[(size_t)(b * NPOINT + m) * NSAMPLE + tid];
  if (tid < 3)       sq[tid]   = newxyz[(size_t)b * 3 * NPOINT + tid * NPOINT + m];
  __syncthreads();

  const float* xb = xyz + (size_t)b * 3 * NPTS;
  const float* fb = features + (size_t)b * NCH * NPTS;

  for (int e = tid; e < 67 * NSAMPLE; e += 256) {
    const int c = e >> 5;          // NSAMPLE == 32
    const int s = e & 31;
    const int idx = sidx[s];
    float v;
    if (c < 3) v = xb[c * NPTS + idx] - sq[c];
    else       v = fb[(c - 3) * NPTS + idx];
    outfeat[(((size_t)b * 67 + c) * NPOINT + m) * NSAMPLE + s] = v;
  }
}

// ---------------------------------------------------------------------------
extern "C" void kernel_launch(void* const* d_in, const int* in_sizes, int n_in,
                              void* d_out, int out_size, void* d_ws, size_t ws_size,
                              hipStream_t stream) {
  const float* xyz      = (const float*)d_in[0];  // (8, 3, 16384)
  const float* features = (const float*)d_in[1];  // (8, 64, 16384)

  float* out        = (float*)d_out;
  float* out_newxyz = out;                               // (8, 3, 1024)
  float* out_feat   = out + (size_t)NBATCH * 3 * NPOINT; // (8, 67, 1024, 32)

  int* wsidx = (int*)d_ws;  // (8, 1024, 32) int32 neighbor indices (1 MB)

  fps_kernel<<<NBATCH, 1024, 0, stream>>>(xyz, out_newxyz);
  ballquery_kernel<<<NBATCH * 64, 32, 0, stream>>>(xyz, out_newxyz, wsidx);
  group_kernel<<<NBATCH * NPOINT, 256, 0, stream>>>(xyz, features, out_newxyz,
                                                    wsidx, out_feat);
}